// VQQuantizer_68753836474701
// MI455X (gfx1250) — compile-verified
//
#include <hip/hip_runtime.h>
#include <math.h>

#define NPTS   1024   // points per group
#define NCENT  256    // centroids
#define KITERS 10
#define TPB    256    // 8 waves (wave32)

typedef __attribute__((ext_vector_type(2))) float v2f;
typedef __attribute__((ext_vector_type(8))) float v8f;

typedef __attribute__((address_space(1))) int gint_t;  // global (AS1)
typedef __attribute__((address_space(3))) int lint_t;  // LDS (AS3)

#if __has_builtin(__builtin_amdgcn_global_load_async_to_lds_b32)
#define HAVE_ASYNC_LDS 1
#else
#define HAVE_ASYNC_LDS 0
#endif

__global__ __launch_bounds__(TPB)
void vq_kmeans_kernel(const float* __restrict__ X, float* __restrict__ out) {
  __shared__ float Xs0[NPTS];
  __shared__ float Xs1[NPTS];
  __shared__ float key[NPTS];
  __shared__ int   ord[NPTS];
  __shared__ float Cs0[NCENT];
  __shared__ float Cs1[NCENT];
  __shared__ float sum0[NCENT];
  __shared__ float sum1[NCENT];
  __shared__ float cnt[NCENT];
  __shared__ float acc[5];
  __shared__ float redMin[TPB];
  __shared__ float redMax[TPB];

  const int tid = threadIdx.x;
  const int g   = blockIdx.x;
  const float* Xg = X + (size_t)g * (NPTS * 2);

  __builtin_prefetch(Xg, 0, 3);  // global_prefetch_b8

  // ---- stage group into LDS (SoA) via async global->LDS DMA when available ----
#if HAVE_ASYNC_LDS
  for (int i = tid; i < NPTS; i += TPB) {
    __builtin_amdgcn_global_load_async_to_lds_b32(
        (gint_t*)(Xg + 2 * i + 0), (lint_t*)&Xs0[i], 0, 0);
    __builtin_amdgcn_global_load_async_to_lds_b32(
        (gint_t*)(Xg + 2 * i + 1), (lint_t*)&Xs1[i], 0, 0);
  }
#if __has_builtin(__builtin_amdgcn_s_wait_asynccnt)
  __builtin_amdgcn_s_wait_asynccnt(0);
#else
  asm volatile("s_wait_asynccnt 0x0" ::: "memory");
#endif
#else
  for (int i = tid; i < NPTS; i += TPB) {
    Xs0[i] = Xg[2 * i + 0];
    Xs1[i] = Xg[2 * i + 1];
  }
#endif
  if (tid < 5) acc[tid] = 0.0f;
  __syncthreads();

  // ---- mean + covariance (block reduction via LDS float atomics) ----
  float s0 = 0.f, s1 = 0.f, s00 = 0.f, s01 = 0.f, s11 = 0.f;
  for (int i = tid; i < NPTS; i += TPB) {
    float a = Xs0[i], b = Xs1[i];
    s0 += a; s1 += b; s00 += a * a; s01 += a * b; s11 += b * b;
  }
  atomicAdd(&acc[0], s0);  atomicAdd(&acc[1], s1);
  atomicAdd(&acc[2], s00); atomicAdd(&acc[3], s01); atomicAdd(&acc[4], s11);
  __syncthreads();

  const float invN = 1.0f / (float)NPTS;
  float mu0 = acc[0] * invN, mu1 = acc[1] * invN;
  float c00 = acc[2] - (float)NPTS * mu0 * mu0;
  float c01 = acc[3] - (float)NPTS * mu0 * mu1;
  float c11 = acc[4] - (float)NPTS * mu1 * mu1;
  float idet = 1.0f / (c00 * c11 - c01 * c01);
  float l00 = c11 * idet, l01 = -c01 * idet, l11 = c00 * idet;

  for (int i = tid; i < NPTS; i += TPB) {
    float dx = Xs0[i] - mu0, dy = Xs1[i] - mu1;
    key[i] = l00 * dx * dx + 2.0f * l01 * dx * dy + l11 * dy * dy;
    ord[i] = i;
  }
  __syncthreads();

  // ---- bitonic ascending sort of (key, ord) in LDS ----
  for (int k = 2; k <= NPTS; k <<= 1) {
    for (int j = k >> 1; j > 0; j >>= 1) {
      for (int t = tid; t < NPTS; t += TPB) {
        int ixj = t ^ j;
        if (ixj > t) {
          float ka = key[t], kb = key[ixj];
          bool up = ((t & k) == 0);
          if (up ? (ka > kb) : (ka < kb)) {
            key[t] = kb; key[ixj] = ka;
            int ia = ord[t]; ord[t] = ord[ixj]; ord[ixj] = ia;
          }
        }
      }
      __syncthreads();
    }
  }

  // ---- select 256 evenly-spaced (by Mahalanobis rank) initial centroids ----
  if (tid < NCENT) {
    int pos = (int)rintf((float)tid * (float)(NPTS - 1) / (float)(NCENT - 1));
    int p = ord[pos];
    Cs0[tid] = Xs0[p];
    Cs1[tid] = Xs1[p];
  }
  __syncthreads();

  const int  lane   = tid & 31;
  const int  wave   = tid >> 5;
  const int  laneN  = lane & 15;
  const bool hiHalf = lane >= 16;

  // ---- k-means iterations: WMMA assignment fused with m-step ----
  for (int it = 0; it < KITERS; ++it) {
    if (tid < NCENT) { sum0[tid] = 0.f; sum1[tid] = 0.f; cnt[tid] = 0.f; }
    __syncthreads();

    // B fragments (4x16, f32): col n = [-2*c0, -2*c1, |c|^2, 0]
    // lanes 0-15 hold K=0,1 ; lanes 16-31 hold K=2,3
    v2f bfrag[16];
#pragma unroll
    for (int t = 0; t < 16; ++t) {
      float cx = Cs0[t * 16 + laneN];
      float cy = Cs1[t * 16 + laneN];
      v2f b;
      b.x = hiHalf ? (cx * cx + cy * cy) : (-2.0f * cx);
      b.y = hiHalf ? 0.0f : (-2.0f * cy);
      bfrag[t] = b;
    }

    // each wave sweeps 8 M-tiles of 16 points
    for (int mt = wave; mt < NPTS / 16; mt += 8) {
      const int p0 = mt * 16;
      // A fragment (16x4, f32): row m = [x0, x1, 1, 0]
      // unconditional LDS reads; select affine constants for the K=2,3 half
      float xa = Xs0[p0 + laneN];
      float xb = Xs1[p0 + laneN];
      v2f a;
      a.x = hiHalf ? 1.0f : xa;
      a.y = hiHalf ? 0.0f : xb;

      float best[8];
      int   bidx[8];
#pragma unroll
      for (int r = 0; r < 8; ++r) { best[r] = 3.4e38f; bidx[r] = 0; }

#pragma unroll
      for (int t = 0; t < 16; ++t) {
        v8f cz = {};
        v8f d = __builtin_amdgcn_wmma_f32_16x16x4_f32(
            false, a, false, bfrag[t], (short)0, cz, false, false);
        const int ci = t * 16 + laneN;
#pragma unroll
        for (int r = 0; r < 8; ++r) {
          float s = d[r];
          bidx[r] = (s < best[r]) ? ci : bidx[r];  // mask consumed immediately
          best[r] = fminf(best[r], s);             // v_min_num_f32
        }
      }

      // argmin over the 16 N-lanes of each half (tie -> lower index)
#pragma unroll
      for (int r = 0; r < 8; ++r) {
#pragma unroll
        for (int off = 8; off >= 1; off >>= 1) {
          float ob = __shfl_xor(best[r], off, 32);
          int   oi = __shfl_xor(bidx[r], off, 32);
          bool take = (ob < best[r]) || (ob == best[r] && oi < bidx[r]);
          bidx[r] = take ? oi : bidx[r];
          best[r] = fminf(best[r], ob);
        }
      }

      // fused m-step scatter: lane0 covers M=r..r+7, lane16 covers M=r+8..r+15
      if (lane == 0 || lane == 16) {
        int rowBase = p0 + (hiHalf ? 8 : 0);
#pragma unroll
        for (int r = 0; r < 8; ++r) {
          int p  = rowBase + r;
          int ci = bidx[r];
          atomicAdd(&sum0[ci], Xs0[p]);   // ds_add_f32
          atomicAdd(&sum1[ci], Xs1[p]);
          atomicAdd(&cnt[ci], 1.0f);
        }
      }
    }
    __syncthreads();

    if (tid < NCENT) {
      float c = fmaxf(cnt[tid], 1.0f);
      Cs0[tid] = sum0[tid] / c;
      Cs1[tid] = sum1[tid] / c;
    }
    __syncthreads();
  }

  // ---- per-group symmetric int8 quantization of centroids ----
  {
    float v0 = (tid & 1) ? Cs1[tid >> 1] : Cs0[tid >> 1];
    int   i1 = tid + TPB;
    float v1 = (i1 & 1) ? Cs1[i1 >> 1] : Cs0[i1 >> 1];
    redMin[tid] = fminf(v0, v1);
    redMax[tid] = fmaxf(v0, v1);
    __syncthreads();
    for (int s = TPB / 2; s > 0; s >>= 1) {
      if (tid < s) {
        redMin[tid] = fminf(redMin[tid], redMin[tid + s]);
        redMax[tid] = fmaxf(redMax[tid], redMax[tid + s]);
      }
      __syncthreads();
    }
    float qmax     = fmaxf(fabsf(redMin[0]), redMax[0]);
    float scale    = qmax / 127.0f;
    float invScale = 127.0f / qmax;
    float* og = out + (size_t)g * (NCENT * 2);
    for (int i = tid; i < NCENT * 2; i += TPB) {
      float v = (i & 1) ? Cs1[i >> 1] : Cs0[i >> 1];
      float q = fminf(fmaxf(rintf(v * invScale), -128.0f), 127.0f) * scale;
      og[i] = q;
    }
  }
}

extern "C" void kernel_launch(void* const* d_in, const int* in_sizes, int n_in,
                              void* d_out, int out_size, void* d_ws, size_t ws_size,
                              hipStream_t stream) {
  const float* X = (const float*)d_in[0];
  float* out = (float*)d_out;
  int groups = in_sizes[0] / (NPTS * 2);   // 1024 for the reference shapes
  vq_kmeans_kernel<<<groups, TPB, 0, stream>>>(X, out);
}